// Net_full_11390253269723
// MI455X (gfx1250) — compile-verified
//
#include <hip/hip_runtime.h>

typedef __attribute__((ext_vector_type(2))) float v2f;
typedef __attribute__((ext_vector_type(8))) float v8f;

// ---------------- elementwise / degree kernels ----------------

__global__ void k_fill(float* p, float v, int n) {
  int i = blockIdx.x * blockDim.x + threadIdx.x;
  if (i < n) p[i] = v;
}

__global__ void k_zero(float* p, int n) {
  int i = blockIdx.x * blockDim.x + threadIdx.x;
  if (i < n) p[i] = 0.0f;
}

__global__ void k_degree(const long long* __restrict__ dst, float* deg, int E) {
  int e = blockIdx.x * blockDim.x + threadIdx.x;
  if (e < E) atomicAdd(&deg[(int)dst[e]], 1.0f);
}

__global__ void k_rsqrt(float* dinv, int n) {
  int i = blockIdx.x * blockDim.x + threadIdx.x;
  if (i < n) dinv[i] = rsqrtf(dinv[i]);   // deg >= 1 always (self-loop)
}

// xw[n][f] = x[n][0]*W1[0][f] + x[n][1]*W1[1][f]   (Fin = 2)
__global__ void k_xw1(const float* __restrict__ x, const float* __restrict__ W1,
                      float* __restrict__ xw, int n) {
  int t = blockIdx.x * blockDim.x + threadIdx.x;
  if (t >= n * 32) return;
  int node = t >> 5, f = t & 31;
  xw[t] = fmaf(x[node * 2], W1[f], x[node * 2 + 1] * W1[32 + f]);
}

// ---------------- edge scatter (the memory-bound core) ----------------
// 8 threads per edge; each handles 4 features (float4 gather, 4 float atomics).
// Edge streams are sequential (HW-prefetch friendly); xw gathers hit L2
// (12.8 MB buffer inside 192 MB L2).
__global__ void k_scatter(const long long* __restrict__ src,
                          const long long* __restrict__ dst,
                          const float* __restrict__ dinv,
                          const float* __restrict__ xw,
                          float* __restrict__ hacc, int E) {
  int t = blockIdx.x * blockDim.x + threadIdx.x;
  if (t >= E * 8) return;
  int e = t >> 3, q = t & 7;
  int s = (int)src[e], d = (int)dst[e];
  float norm = dinv[s] * dinv[d];
  float4 v = ((const float4*)(xw + (size_t)s * 32))[q];
  float* hd = hacc + (size_t)d * 32 + q * 4;
  atomicAdd(hd + 0, v.x * norm);
  atomicAdd(hd + 1, v.y * norm);
  atomicAdd(hd + 2, v.z * norm);
  atomicAdd(hd + 3, v.w * norm);
}

// h = relu(hacc + xw * dinv^2 (self-loop) + bias)
__global__ void k_epilogue(const float* __restrict__ xw, const float* __restrict__ dinv,
                           const float* __restrict__ b, float* __restrict__ h, int n) {
  int t = blockIdx.x * blockDim.x + threadIdx.x;
  if (t >= n * 32) return;
  int node = t >> 5, f = t & 31;
  float di = dinv[node];
  float v = h[t] + xw[t] * di * di + b[f];
  h[t] = fmaxf(v, 0.0f);
}

// ---------------- 32x32 GEMM via V_WMMA_F32_16X16X4_F32 ----------------
// hout[r][c] = sum_k hin[r][k]*W[k][c] (+bias, optional relu)
// One wave32 per 16-row tile; 8 K-steps of 4; two 16-col N tiles.
// A frag (16x4 f32): lane l holds row m=l&15; vgpr v holds K = (l>>4)*2 + v.
// B frag (4x16 f32): lane l holds col n=l&15; vgpr v holds K = (l>>4)*2 + v.
// C/D (16x16 f32, 8 vgprs): lane l, vgpr v -> row (l>>4)*8+v, col l&15.
// Rows are clamped (not branch-guarded) so the unrolled A loads stay
// straight-line global_load_b64 with immediate offsets; out-of-range rows
// produce garbage accumulator rows that are masked at the store.
__global__ void __launch_bounds__(256)
k_gemm32_wmma(const float* __restrict__ hin, const float* __restrict__ W,
              const float* __restrict__ bias, float* __restrict__ hout,
              int n, int do_relu) {
  __shared__ float Wlds[32 * 32];
  int tid = threadIdx.x;
  ((float4*)Wlds)[tid] = ((const float4*)W)[tid];  // 256 threads x 4 floats = 1024
  __syncthreads();

  int wave = tid >> 5, lane = tid & 31;
  int tile = blockIdx.x * 8 + wave;
  int r0 = tile * 16;
  if (r0 >= n) return;                 // uniform per wave -> EXEC all-1s inside

  int m = lane & 15;
  int khalf = lane >> 4;

  int arow = r0 + m;
  if (arow >= n) arow = n - 1;         // clamp: keeps loads unconditional
  // A elements for K-step kk live at k = kk*4 + khalf*2 (v2f stride of 2)
  const v2f* ap = (const v2f*)(hin + (size_t)arow * 32 + khalf * 2);

  v8f acc0 = {};
  v8f acc1 = {};
#pragma unroll
  for (int kk = 0; kk < 8; ++kk) {
    v2f a = ap[kk * 2];
    int k0 = kk * 4 + khalf * 2;
    v2f b0, b1;
    b0.x = Wlds[k0 * 32 + m];
    b0.y = Wlds[(k0 + 1) * 32 + m];
    b1.x = Wlds[k0 * 32 + 16 + m];
    b1.y = Wlds[(k0 + 1) * 32 + 16 + m];
    acc0 = __builtin_amdgcn_wmma_f32_16x16x4_f32(false, a, false, b0, (short)0,
                                                 acc0, false, false);
    acc1 = __builtin_amdgcn_wmma_f32_16x16x4_f32(false, a, false, b1, (short)0,
                                                 acc1, false, false);
  }

  int mg = (lane >> 4) * 8;
#pragma unroll
  for (int v = 0; v < 8; ++v) {
    int row = r0 + mg + v;
    if (row < n) {
      float o0 = acc0[v];
      float o1 = acc1[v];
      if (bias) { o0 += bias[m]; o1 += bias[16 + m]; }
      if (do_relu) { o0 = fmaxf(o0, 0.0f); o1 = fmaxf(o1, 0.0f); }
      hout[(size_t)row * 32 + m] = o0;
      hout[(size_t)row * 32 + 16 + m] = o1;
    }
  }
}

// out[n] = dot(f1[n], Wf2) + bf2 — one wave per node, shuffle reduction
__global__ void k_fc2(const float* __restrict__ f1, const float* __restrict__ Wf2,
                      const float* __restrict__ bf2, float* __restrict__ out, int n) {
  int tid = threadIdx.x;
  int wave = tid >> 5, lane = tid & 31;
  int node = blockIdx.x * 8 + wave;
  if (node >= n) return;
  float v = f1[(size_t)node * 32 + lane] * Wf2[lane];
  for (int off = 16; off > 0; off >>= 1) v += __shfl_xor(v, off, 32);
  if (lane == 0) out[node] = v + bf2[0];
}

// ---------------- launcher ----------------

extern "C" void kernel_launch(void* const* d_in, const int* in_sizes, int n_in,
                              void* d_out, int out_size, void* d_ws, size_t ws_size,
                              hipStream_t stream) {
  const float* x        = (const float*)d_in[0];
  const long long* ei   = (const long long*)d_in[1];
  const float* W1       = (const float*)d_in[2];
  const float* b1       = (const float*)d_in[3];
  const float* W2       = (const float*)d_in[4];
  const float* b2       = (const float*)d_in[5];
  const float* Wf1      = (const float*)d_in[6];
  const float* bf1      = (const float*)d_in[7];
  const float* Wf2      = (const float*)d_in[8];
  const float* bf2      = (const float*)d_in[9];
  float* out = (float*)d_out;

  int n = in_sizes[0] / 2;        // 100000 nodes (Fin = 2)
  int E = in_sizes[1] / 2;        // 3.2M edges
  const long long* src = ei;
  const long long* dst = ei + E;

  // workspace: [dinv: n][bufA: n*32][bufB: n*32]  (~26 MB)
  float* dinv = (float*)d_ws;
  float* bufA = dinv + n;
  float* bufB = bufA + (size_t)n * 32;

  const int TB = 256;
  int nf = n * 32;                       // 3.2M
  dim3 gN((n + TB - 1) / TB);
  dim3 gE((E + TB - 1) / TB);
  dim3 gNF((nf + TB - 1) / TB);
  dim3 gE8((E * 8 + TB - 1) / TB);
  int tiles = (n + 15) / 16;
  dim3 gT((tiles + 7) / 8);
  dim3 gW8((n + 7) / 8);

  // ---- degrees: dinv = rsqrt(1 + indegree) ----
  k_fill<<<gN, TB, 0, stream>>>(dinv, 1.0f, n);
  k_degree<<<gE, TB, 0, stream>>>(dst, dinv, E);
  k_rsqrt<<<gN, TB, 0, stream>>>(dinv, n);

  // ---- GCN layer 1 ----
  k_xw1<<<gNF, TB, 0, stream>>>(x, W1, bufA, n);
  k_zero<<<gNF, TB, 0, stream>>>(bufB, nf);
  k_scatter<<<gE8, TB, 0, stream>>>(src, dst, dinv, bufA, bufB, E);
  k_epilogue<<<gNF, TB, 0, stream>>>(bufA, dinv, b1, bufB, n);   // h1 in bufB

  // ---- GCN layer 2 ----
  k_gemm32_wmma<<<gT, TB, 0, stream>>>(bufB, W2, nullptr, bufA, n, 0);  // xw2 in bufA
  k_zero<<<gNF, TB, 0, stream>>>(bufB, nf);
  k_scatter<<<gE8, TB, 0, stream>>>(src, dst, dinv, bufA, bufB, E);
  k_epilogue<<<gNF, TB, 0, stream>>>(bufA, dinv, b2, bufB, n);   // h2 in bufB

  // ---- MLP head ----
  k_gemm32_wmma<<<gT, TB, 0, stream>>>(bufB, Wf1, bf1, bufA, n, 1);     // f1 in bufA
  k_fc2<<<gW8, TB, 0, stream>>>(bufA, Wf2, bf2, out, n);
}